// WeldonPool2d_30477087932836
// MI455X (gfx1250) — compile-verified
//
#include <hip/hip_runtime.h>
#include <stdint.h>

// WeldonPool2d: per (b,c) row of n=H*W floats, out = (mean(top10)+mean(bot10))/2.
// Memory-bound selection (256 MB in / 256 KB out -> ~11us HBM floor @ 23.3 TB/s).
// No matrix contraction exists -> WMMA inapplicable. CDNA5 paths used instead:
//   * global_load_async_to_lds_b128 (ASYNCcnt) for coalesced streaming + LDS
//     transpose so each wave32 lane owns a full row (no cross-lane top-k merge)
//   * split wait counters (s_wait_asynccnt / s_wait_dscnt) for a 3-deep
//     software pipeline
//   * v_med3_num_f32 insertion network: for a sorted list, the branchless
//     top-k insert max(min(prev,x),cur) == median(prev,x,cur) -> 10 VALU ops
//     per insert instead of 19, halving the dominant VALU cost.

#define WAVE_SZ 32
#define WAVES_PER_BLK 8
#define BLK_THREADS (WAVE_SZ * WAVES_PER_BLK)         // 256 threads = 8 wave32s
#define ROWS_PER_WAVE 32                              // one row per lane
#define ROWS_PER_BLK (WAVES_PER_BLK * ROWS_PER_WAVE)  // 256 rows per block
#define COLS_PER_CHUNK 16                             // columns staged per step
#define LDS_STRIDE 20    // 16 data + 4 pad floats: rows stay 16B aligned and
                         // 20*l mod 64 gives uniform 2-way banking (optimal b128)
#define NBUF 3           // triple-buffered pipeline, prefetch distance 2
#define K 10

typedef __attribute__((address_space(3))) float lds_float;

// 32-bit LDS byte address (workgroup-LDS-relative) for the async copy VDST.
static __device__ __forceinline__ uint32_t lds_addr32(const float* p) {
    return (uint32_t)(uintptr_t)(lds_float*)p;
}

// Branchless insert into descending top-K list. Invariant t[i] <= t[i-1], so
//   new[i] = max(t[i], min(t[i-1], x)) == med3(t[i-1], x, t[i])  (1 VALU op).
static __device__ __forceinline__ void ins_top(float (&t)[K], float x) {
    float prev = t[0];
    t[0] = fmaxf(t[0], x);
#pragma unroll
    for (int i = 1; i < K; ++i) {
        float cur = t[i];
        t[i] = __builtin_amdgcn_fmed3f(prev, x, cur);
        prev = cur;
    }
}

// Ascending bottom-K list. Invariant b[i] >= b[i-1], and
//   new[i] = min(b[i], max(b[i-1], x)) == med3(b[i-1], x, b[i]) as well.
static __device__ __forceinline__ void ins_bot(float (&b)[K], float x) {
    float prev = b[0];
    b[0] = fminf(b[0], x);
#pragma unroll
    for (int i = 1; i < K; ++i) {
        float cur = b[i];
        b[i] = __builtin_amdgcn_fmed3f(prev, x, cur);
        prev = cur;
    }
}

__global__ __launch_bounds__(BLK_THREADS)
void weldon_topk_kernel(const float* __restrict__ x, float* __restrict__ out,
                        int rows, int n) {
    // 3 * 8 * 32 * 20 * 4B = 60 KB staging LDS.
    __shared__ float tile[NBUF][WAVES_PER_BLK][ROWS_PER_WAVE][LDS_STRIDE];

    const int lane  = threadIdx.x & (WAVE_SZ - 1);
    const int wave  = threadIdx.x >> 5;
    const int row0  = blockIdx.x * ROWS_PER_BLK + wave * ROWS_PER_WAVE;
    const int myrow = row0 + lane;

    // Wave-uniform 64-bit base (SGPR pair) for the GVS-form async loads.
    const float* base = x + (size_t)row0 * (size_t)n;

    // Chunk = 32 rows x 16 cols = 2 KB = 4 async b128 ops. Instruction t covers
    // rows 8t..8t+7; consecutive lanes take consecutive 16B quads (coalesced).
    const int a_ri = lane >> 2;
    const int a_q  = lane & 3;
    const int nch  = n / COLS_PER_CHUNK;

    // Stage chunk c into buffer p. Async LDS writes are unordered vs DS reads,
    // and buffer p was read by DS loads in the previous iteration -> drain DScnt first.
    auto stage = [&](int c, int p) {
        asm volatile("s_wait_dscnt 0" ::: "memory");
#pragma unroll
        for (int t = 0; t < 4; ++t) {
            int ri = t * 8 + a_ri;
            int gr = row0 + ri;
            gr = (gr < rows) ? gr : (rows - 1);   // tail clamp (no-op for 65536)
            uint32_t voff = (uint32_t)((((gr - row0) * n) + c * COLS_PER_CHUNK + a_q * 4) * 4);
            uint32_t ldst = lds_addr32(&tile[p][wave][ri][a_q * 4]);
            // CDNA5 async copy, GVS mode; streamed once -> non-temporal hint
            // keeps the 256 MB pass from thrashing the 192 MB L2.
            asm volatile("global_load_async_to_lds_b128 %0, %1, %2 th:TH_LOAD_NT"
                         :: "v"(ldst), "v"(voff), "s"(base)
                         : "memory");
        }
    };

    float top[K], bot[K];
#pragma unroll
    for (int i = 0; i < K; ++i) { top[i] = -INFINITY; bot[i] = INFINITY; }

    if (nch > 0) {
        stage(0, 0);
        stage((1 < nch) ? 1 : 0, 1);              // prologue depth 2
        for (int c = 0; c < nch; ++c) {
            const int p = c % NBUF;
            int cpre = c + 2;                     // prefetch distance 2
            const int ppre = cpre % NBUF;
            if (cpre >= nch) cpre = nch - 1;      // tail: harmless redundant refill
            stage(cpre, ppre);
            // 12 outstanding async ops (c, c+1, c+2); in-order completion =>
            // waiting down to <=8 guarantees chunk c's 4 copies have landed.
            asm volatile("s_wait_asynccnt 8" ::: "memory");
#pragma unroll
            for (int j = 0; j < COLS_PER_CHUNK / 4; ++j) {
                // Transposed read-out: lane reads its own row; 16B-aligned,
                // uniformly 2-way banked ds_load_b128.
                const float4 v = *(const float4*)&tile[p][wave][lane][j * 4];
                ins_top(top, v.x); ins_bot(bot, v.x);
                ins_top(top, v.y); ins_bot(bot, v.y);
                ins_top(top, v.z); ins_bot(bot, v.z);
                ins_top(top, v.w); ins_bot(bot, v.w);
            }
        }
    }

    // Generic remainder path (n % 16 != 0); not taken for n = 1024.
    for (int col = nch * COLS_PER_CHUNK; col < n; ++col) {
        if (myrow < rows) {
            float v = x[(size_t)myrow * (size_t)n + col];
            ins_top(top, v); ins_bot(bot, v);
        }
    }

    float sT = 0.f, sB = 0.f;
#pragma unroll
    for (int i = 0; i < K; ++i) { sT += top[i]; sB += bot[i]; }

    // (sT/10 + sB/10) / 2 == (sT + sB) * 0.05 ; coalesced b32 store.
    if (myrow < rows) out[myrow] = (sT + sB) * 0.05f;
}

extern "C" void kernel_launch(void* const* d_in, const int* in_sizes, int n_in,
                              void* d_out, int out_size, void* d_ws, size_t ws_size,
                              hipStream_t stream) {
    const float* x = (const float*)d_in[0];
    float* out = (float*)d_out;

    const int rows = out_size;                        // B*C = 65536
    const long long total = (long long)in_sizes[0];
    const int n = (int)(total / (long long)rows);     // H*W = 1024

    const int blocks = (rows + ROWS_PER_BLK - 1) / ROWS_PER_BLK;   // 256
    weldon_topk_kernel<<<blocks, BLK_THREADS, 0, stream>>>(x, out, rows, n);
}